// Voxurf_59193239274074
// MI455X (gfx1250) — compile-verified
//
#include <hip/hip_runtime.h>

typedef float v2f __attribute__((ext_vector_type(2)));
typedef float v8f __attribute__((ext_vector_type(8)));

#define THREADS 256
#define WAVE 32
#define TILE 256                 // elements scanned per wave via WMMA
#define BLOCK_ELEMS 2048         // 8 waves * 256

__device__ __forceinline__ float clipa(float a) {
    return fminf(fmaxf(a, 0.0f), 1.0f - 1e-7f);
}

__device__ __forceinline__ float rdlane(float v, int l) {
    return __int_as_float(__builtin_amdgcn_readlane(__float_as_int(v), l));
}

// ---------------- K1: per-block sum of logs ----------------
__global__ void k1_blocksum(const float* __restrict__ alpha, float* __restrict__ bs,
                            long long M) {
    __shared__ float sh[THREADS];
    const int t = threadIdx.x;
    const long long base = (long long)blockIdx.x * BLOCK_ELEMS;
    float s = 0.0f;
#pragma unroll
    for (int k = 0; k < 8; ++k) {
        long long i = base + (long long)k * THREADS + t;
        if (i < M) s += log1pf(-clipa(alpha[i]));
    }
    sh[t] = s;
    __syncthreads();
    for (int off = THREADS / 2; off > 0; off >>= 1) {
        if (t < off) sh[t] += sh[t + off];
        __syncthreads();
    }
    if (t == 0) bs[blockIdx.x] = sh[0];
}

// ---------------- K2: exclusive scan of block sums (1 workgroup) ----------------
__global__ void k2_scan(const float* __restrict__ bs, float* __restrict__ bp, int nb) {
    __shared__ float sh[THREADS];
    const int t = threadIdx.x;
    const int per = (nb + THREADS - 1) / THREADS;
    const int beg = t * per;
    const int end = min(beg + per, nb);
    float s = 0.0f;
    for (int i = beg; i < end; ++i) s += bs[i];
    sh[t] = s;
    __syncthreads();
    for (int off = 1; off < THREADS; off <<= 1) {
        float u = (t >= off) ? sh[t - off] : 0.0f;
        __syncthreads();
        sh[t] += u;
        __syncthreads();
    }
    float run = sh[t] - s;          // exclusive base for this thread's range
    for (int i = beg; i < end; ++i) { bp[i] = run; run += bs[i]; }
    if (t == THREADS - 1) bp[nb] = sh[THREADS - 1];   // grand total
}

// ---------------- K3: global exclusive cumsum of logs, WMMA tile scan ----------------
// Exclusive scan of 256 elems per wave: S = X * U_strict (16x16x16 product via four
// chained V_WMMA_F32_16X16X4_F32) + per-row scalar prefix broadcast.
__global__ void k3_scan(const float* __restrict__ alpha, const float* __restrict__ bp,
                        float* __restrict__ excl, long long M) {
    __shared__ float waveTot[8];
    const int lane = threadIdx.x & (WAVE - 1);
    const int wave = threadIdx.x >> 5;
    const int half = lane >> 4;        // 0: lanes 0-15, 1: lanes 16-31
    const int mrow = lane & 15;        // A: M row; B/C: N column
    const long long tileBase = (long long)blockIdx.x * BLOCK_ELEMS + (long long)wave * TILE;

    // Gather logs in f32 A-matrix layout (chunk q covers K columns 4q..4q+3):
    // lanes 0-15 hold K = 4q+0, 4q+1; lanes 16-31 hold K = 4q+2, 4q+3.
    float xa0[4], xa1[4];
#pragma unroll
    for (int q = 0; q < 4; ++q) {
        const int c0 = 4 * q + 2 * half;
        const long long i0 = tileBase + (long long)mrow * 16 + c0;
        float a0 = (i0 < M) ? alpha[i0] : 0.0f;       // alpha=0 -> log term 0 (identity)
        float a1 = (i0 + 1 < M) ? alpha[i0 + 1] : 0.0f;
        xa0[q] = log1pf(-clipa(a0));
        xa1[q] = log1pf(-clipa(a1));
    }

    // D = X * U_strict (U_strict[k][n] = 1 if k < n): exclusive scan along each row.
    v8f acc = {0.f, 0.f, 0.f, 0.f, 0.f, 0.f, 0.f, 0.f};
#pragma unroll
    for (int q = 0; q < 4; ++q) {
        const int k0 = 4 * q + 2 * half;   // B rows held by this half-wave for chunk q
        v2f A = {xa0[q], xa1[q]};
        v2f B = {(k0 < mrow) ? 1.0f : 0.0f, (k0 + 1 < mrow) ? 1.0f : 0.0f};
        acc = __builtin_amdgcn_wmma_f32_16x16x4_f32(
            /*neg_a=*/false, A, /*neg_b=*/false, B,
            /*c_mod=*/(short)0, acc, /*reuse_a=*/false, /*reuse_b=*/false);
    }

    // Row totals rt[r] = S_excl[r][15] + X[r][15]  (X[r][15] lives in xa1[3] on lane 16+r).
    float rt[16];
#pragma unroll
    for (int r = 0; r < 8; ++r) rt[r] = rdlane(acc[r], 15) + rdlane(xa1[3], 16 + r);
#pragma unroll
    for (int r = 0; r < 8; ++r) rt[r + 8] = rdlane(acc[r], 31) + rdlane(xa1[3], 24 + r);

    // Scalar exclusive prefix over the 16 row totals (uniform -> SALU/SGPR math).
    float P[16];
    float run = 0.0f;
#pragma unroll
    for (int r = 0; r < 16; ++r) { P[r] = run; run += rt[r]; }
    const float tileTotal = run;

    // Broadcast row prefixes into the C-layout result (VGPR v: row v / v+8).
#pragma unroll
    for (int v = 0; v < 8; ++v) acc[v] += half ? P[v + 8] : P[v];

    if (lane == 0) waveTot[wave] = tileTotal;
    __syncthreads();
    float wbase = 0.0f;
    for (int w = 0; w < wave; ++w) wbase += waveTot[w];
    const float base = bp[blockIdx.x] + wbase;

    // Store: lane<16 VGPR v = element (v*16 + lane); lane>=16 = ((v+8)*16 + lane-16).
#pragma unroll
    for (int v = 0; v < 8; ++v) {
        const int row = v + 8 * half;
        const long long e = tileBase + (long long)row * 16 + mrow;
        if (e < M) excl[e] = base + acc[v];
    }
}

// ---------------- K4a: fill per-ray segment prefixes E[0..N] ----------------
__global__ void k4a_fill(const int* __restrict__ ray, const float* __restrict__ excl,
                         const float* __restrict__ bp, float* __restrict__ E,
                         long long M, int N, int nb) {
    const long long i = (long long)blockIdx.x * THREADS + threadIdx.x;
    if (i >= M) return;
    const int r = ray[i];
    const int rp = (i == 0) ? -1 : ray[i - 1];
    if (r != rp) {
        const float ex = excl[i];
        for (int q = rp + 1; q <= r; ++q) E[q] = ex;   // covers empty rays in the gap
    }
    if (i == M - 1) {
        const float tot = bp[nb];
        for (int q = r + 1; q <= N; ++q) E[q] = tot;   // trailing empty rays + E[N]
    }
}

// ---------------- K4b: weights = alpha * exp(segment-local exclusive scan) ----------------
__global__ void k4b_weights(const float* __restrict__ alpha, const int* __restrict__ ray,
                            const float* __restrict__ excl, const float* __restrict__ E,
                            float* __restrict__ outw, long long M) {
    const long long i = (long long)blockIdx.x * THREADS + threadIdx.x;
    if (i >= M) return;
    const float a = clipa(alpha[i]);
    outw[i] = a * expf(excl[i] - E[ray[i]]);
}

// ---------------- K5: alphainv_last = exp(segment sum) ----------------
__global__ void k5_alphainv(const float* __restrict__ E, float* __restrict__ outA, int N) {
    const int r = blockIdx.x * THREADS + threadIdx.x;
    if (r < N) outA[r] = expf(E[r + 1] - E[r]);
}

extern "C" void kernel_launch(void* const* d_in, const int* in_sizes, int n_in,
                              void* d_out, int out_size, void* d_ws, size_t ws_size,
                              hipStream_t stream) {
    const float* alpha = (const float*)d_in[0];
    const int* ray = (const int*)d_in[1];
    const long long M = (long long)in_sizes[0];
    const int N = out_size - (int)M;                 // output = weights[M] ++ alphainv[N]
    const long long nb = (M + BLOCK_ELEMS - 1) / BLOCK_ELEMS;

    float* ws = (float*)d_ws;
    float* excl = ws;                    // M
    float* bs = excl + M;                // nb
    float* bp = bs + nb;                 // nb + 1
    float* E = bp + nb + 1;              // N + 1

    float* outw = (float*)d_out;
    float* outA = outw + M;

    const dim3 blk(THREADS);
    const unsigned gScan = (unsigned)nb;
    const unsigned gLin = (unsigned)((M + THREADS - 1) / THREADS);
    const unsigned gRay = (unsigned)((N + THREADS - 1) / THREADS);

    k1_blocksum<<<gScan, blk, 0, stream>>>(alpha, bs, M);
    k2_scan<<<1, blk, 0, stream>>>(bs, bp, (int)nb);
    k3_scan<<<gScan, blk, 0, stream>>>(alpha, bp, excl, M);
    k4a_fill<<<gLin, blk, 0, stream>>>(ray, excl, bp, E, M, N, (int)nb);
    k4b_weights<<<gLin, blk, 0, stream>>>(alpha, ray, excl, E, outw, M);
    k5_alphainv<<<gRay, blk, 0, stream>>>(E, outA, N);
}